// FCGF_point_att2_ican_fc_89575837925674
// MI455X (gfx1250) — compile-verified
//
#include <hip/hip_runtime.h>
#include <hip/hip_bf16.h>

typedef __attribute__((ext_vector_type(2))) float v2f;
typedef __attribute__((ext_vector_type(8))) float v8f;

#define N_TOTAL (1 << 20)
#define D_IN    32
#define B_SEG   64
#define SEG_LEN 16384
#define D_OUT   64
#define EPS     1e-5f

// workspace layout in floats
#define WS_ZM    0                  // float2[N]      (8 MB)
#define WS_E     (2 * N_TOTAL)      // float [N]      (4 MB)
#define WS_BN1P  (3 * N_TOTAL)      // float2[4096]   (32 KB)
#define WS_SCAL  (WS_BN1P + 8192)   // [0]=mu [1]=g1*rstd
#define WS_COEF  (WS_SCAL + 16)     // float [64]
#define WS_POOLP (WS_COEF + 64)     // float [512*32] (64 KB)

// ---------- k1: per-row z = x.w1 + b1, m = sum(x); global BN1 partial sums ----------
__global__ void k1_row_stats(const float* __restrict__ x,
                             const float* __restrict__ w1,
                             const float* __restrict__ b1p,
                             float2* __restrict__ zm,
                             float2* __restrict__ bn1p) {
    __shared__ float2 red[256];
    int i = blockIdx.x * 256 + threadIdx.x;
    const float4* xr = (const float4*)(x + (size_t)i * D_IN);
    const float4* wv = (const float4*)w1;
    float z = b1p[0], m = 0.f;
#pragma unroll
    for (int q = 0; q < 8; ++q) {
        float4 xv = xr[q];
        float4 wq = wv[q];
        z = fmaf(xv.x, wq.x, z); z = fmaf(xv.y, wq.y, z);
        z = fmaf(xv.z, wq.z, z); z = fmaf(xv.w, wq.w, z);
        m += xv.x + xv.y + xv.z + xv.w;
    }
    zm[i] = make_float2(z, m);
    red[threadIdx.x] = make_float2(z, z * z);
    __syncthreads();
    for (int s = 128; s > 0; s >>= 1) {
        if (threadIdx.x < s) {
            float2 o = red[threadIdx.x + s];
            red[threadIdx.x].x += o.x;
            red[threadIdx.x].y += o.y;
        }
        __syncthreads();
    }
    if (threadIdx.x == 0) bn1p[blockIdx.x] = red[0];
}

// ---------- k2: fold 4096 partials -> mu, a = g1 * rsqrt(var+eps) ----------
__global__ void k2_bn1_finalize(const float2* __restrict__ bn1p,
                                const float* __restrict__ g1,
                                float* __restrict__ scal) {
    __shared__ float2 red[1024];
    int t = threadIdx.x;
    float s1 = 0.f, s2 = 0.f;
#pragma unroll
    for (int q = 0; q < 4; ++q) {
        float2 p = bn1p[t * 4 + q];
        s1 += p.x; s2 += p.y;
    }
    red[t] = make_float2(s1, s2);
    __syncthreads();
    for (int s = 512; s > 0; s >>= 1) {
        if (t < s) { red[t].x += red[t + s].x; red[t].y += red[t + s].y; }
        __syncthreads();
    }
    if (t == 0) {
        float mu  = red[0].x / (float)N_TOTAL;
        float var = red[0].y / (float)N_TOTAL - mu * mu;
        scal[0] = mu;
        scal[1] = g1[0] * rsqrtf(var + EPS);
    }
}

// ---------- k3: per-segment softmax: e = exp(s - max), coef = 1/(sum*len) ----------
__global__ void k3_softmax(const float2* __restrict__ zm,
                           const float* __restrict__ scal,
                           const float* __restrict__ beta1p,
                           float* __restrict__ E,
                           float* __restrict__ coef) {
    __shared__ float red[512];
    int seg = blockIdx.x, t = threadIdx.x;
    float mu = scal[0], a = scal[1], beta1 = beta1p[0];
    int base = seg * SEG_LEN;
    float lmax = -3.402823466e38f;
    for (int k = 0; k < SEG_LEN / 512; ++k) {
        int i = base + k * 512 + t;
        float2 p = zm[i];
        float s = (a * (p.x - mu) + beta1) * (p.y * (1.0f / 32.0f));
        E[i] = s;
        lmax = fmaxf(lmax, s);
    }
    red[t] = lmax;
    __syncthreads();
    for (int s = 256; s > 0; s >>= 1) {
        if (t < s) red[t] = fmaxf(red[t], red[t + s]);
        __syncthreads();
    }
    float smax = red[0];
    __syncthreads();
    float lsum = 0.f;
    for (int k = 0; k < SEG_LEN / 512; ++k) {
        int i = base + k * 512 + t;
        float e = expf(E[i] - smax);
        E[i] = e;
        lsum += e;
    }
    red[t] = lsum;
    __syncthreads();
    for (int s = 256; s > 0; s >>= 1) {
        if (t < s) red[t] += red[t + s];
        __syncthreads();
    }
    if (t == 0) coef[seg] = 1.0f / (red[0] * (float)SEG_LEN);
}

// ---------- k4: weighted pooling partials: 8 blocks per segment ----------
__global__ void k4_pool(const float* __restrict__ x,
                        const float* __restrict__ E,
                        float* __restrict__ poolp) {
    __shared__ float acc1[256 * 33]; // stride-33 padding: conflict-free
    __shared__ float acc2[8 * 32];
    int seg = blockIdx.x >> 3, sub = blockIdx.x & 7, t = threadIdx.x;
    int base = seg * SEG_LEN + sub * 2048;
    float acc[32];
#pragma unroll
    for (int j = 0; j < 32; ++j) acc[j] = 0.f;
    for (int r = 0; r < 8; ++r) {
        int i = base + r * 256 + t;
        float e = E[i];
        const float4* xr = (const float4*)(x + (size_t)i * D_IN);
#pragma unroll
        for (int q = 0; q < 8; ++q) {
            float4 xv = xr[q];
            acc[q * 4 + 0] = fmaf(e, xv.x, acc[q * 4 + 0]);
            acc[q * 4 + 1] = fmaf(e, xv.y, acc[q * 4 + 1]);
            acc[q * 4 + 2] = fmaf(e, xv.z, acc[q * 4 + 2]);
            acc[q * 4 + 3] = fmaf(e, xv.w, acc[q * 4 + 3]);
        }
    }
#pragma unroll
    for (int j = 0; j < 32; ++j) acc1[t * 33 + j] = acc[j];
    __syncthreads();
    int c = t >> 5, j = t & 31;
    float p = 0.f;
    for (int r = 0; r < 32; ++r) p += acc1[(c * 32 + r) * 33 + j];
    acc2[c * 32 + j] = p;
    __syncthreads();
    if (t < 32) {
        float q = 0.f;
#pragma unroll
        for (int cc = 0; cc < 8; ++cc) q += acc2[cc * 32 + t];
        poolp[blockIdx.x * 32 + t] = q;
    }
}

// ---------- k5: pooled@W2+b2 via V_WMMA_F32_16X16X4_F32, then BN2 ----------
__global__ void k5_fc_bn2(const float* __restrict__ poolp,
                          const float* __restrict__ coef,
                          const float* __restrict__ W2,
                          const float* __restrict__ b2,
                          const float* __restrict__ g2,
                          const float* __restrict__ beta2,
                          float* __restrict__ out) {
    __shared__ float pooled[64 * 32];
    __shared__ float w2s[32 * 64];
    __shared__ float hbuf[64 * 64];
    __shared__ float colScale[64], colShift[64];
    int t = threadIdx.x; // 512 threads = 16 waves
    // fold 8 pooling partials per segment (fixed order) and scale by coef
#pragma unroll
    for (int k = 0; k < 4; ++k) {
        int idx = k * 512 + t;      // idx = b*32 + j
        int b = idx >> 5, j = idx & 31;
        float s = 0.f;
#pragma unroll
        for (int p = 0; p < 8; ++p) s += poolp[(b * 8 + p) * 32 + j];
        pooled[idx] = s * coef[b];
    }
#pragma unroll
    for (int k = 0; k < 4; ++k) {
        int idx = k * 512 + t;
        w2s[idx] = W2[idx];
    }
    __syncthreads();

    int wv = t >> 5, lane = t & 31;
    int tm = wv >> 2, tn = wv & 3;          // 4x4 grid of 16x16 tiles
    int hi = (lane >> 4) & 1, lr = lane & 15;
    float bias = b2[tn * 16 + lr];
    v8f c;
#pragma unroll
    for (int r = 0; r < 8; ++r) c[r] = bias;
#pragma unroll
    for (int k = 0; k < 8; ++k) {           // K = 32 in steps of 4
        int kk = k * 4;
        v2f av, bv;
        int arow = tm * 16 + lr;
        av[0] = pooled[arow * 32 + kk + (hi ? 2 : 0)];
        av[1] = pooled[arow * 32 + kk + (hi ? 3 : 1)];
        bv[0] = w2s[(kk + (hi ? 2 : 0)) * 64 + tn * 16 + lr];
        bv[1] = w2s[(kk + (hi ? 3 : 1)) * 64 + tn * 16 + lr];
        c = __builtin_amdgcn_wmma_f32_16x16x4_f32(
                false, av, false, bv, (short)0, c, false, false);
    }
#pragma unroll
    for (int r = 0; r < 8; ++r) {
        int rowm = tm * 16 + (hi ? 8 + r : r);
        hbuf[rowm * 64 + tn * 16 + lr] = c[r];
    }
    __syncthreads();

    if (t < 64) { // BN2 over the 64 rows, per column, fixed order
        float s1 = 0.f, s2 = 0.f;
        for (int i = 0; i < 64; ++i) {
            float h = hbuf[i * 64 + t];
            s1 += h; s2 += h * h;
        }
        float mu2  = s1 * (1.0f / 64.0f);
        float var2 = s2 * (1.0f / 64.0f) - mu2 * mu2;
        float sc   = g2[t] * rsqrtf(var2 + EPS);
        colScale[t] = sc;
        colShift[t] = beta2[t] - sc * mu2;
    }
    __syncthreads();
#pragma unroll
    for (int k = 0; k < 8; ++k) {
        int idx = k * 512 + t;
        int jc = idx & 63;
        out[idx] = hbuf[idx] * colScale[jc] + colShift[jc];
    }
}

extern "C" void kernel_launch(void* const* d_in, const int* in_sizes, int n_in,
                              void* d_out, int out_size, void* d_ws, size_t ws_size,
                              hipStream_t stream) {
    const float* x     = (const float*)d_in[0];
    // d_in[1] = length[64] (all SEG_LEN, static shapes) -- unused
    const float* w1    = (const float*)d_in[2];
    const float* b1    = (const float*)d_in[3];
    const float* g1    = (const float*)d_in[4];
    const float* beta1 = (const float*)d_in[5];
    const float* W2    = (const float*)d_in[6];
    const float* b2    = (const float*)d_in[7];
    const float* g2    = (const float*)d_in[8];
    const float* beta2 = (const float*)d_in[9];

    float*  ws   = (float*)d_ws;
    float2* zm   = (float2*)(ws + WS_ZM);
    float*  E    = ws + WS_E;
    float2* bn1p = (float2*)(ws + WS_BN1P);
    float*  scal = ws + WS_SCAL;
    float*  coef = ws + WS_COEF;
    float*  poolp= ws + WS_POOLP;

    k1_row_stats  <<<N_TOTAL / 256, 256, 0, stream>>>(x, w1, b1, zm, bn1p);
    k2_bn1_finalize<<<1, 1024, 0, stream>>>(bn1p, g1, scal);
    k3_softmax    <<<B_SEG, 512, 0, stream>>>(zm, scal, beta1, E, coef);
    k4_pool       <<<B_SEG * 8, 256, 0, stream>>>(x, E, poolp);
    k5_fc_bn2     <<<1, 512, 0, stream>>>(poolp, coef, W2, b2, g2, beta2, (float*)d_out);
}